// DetectPostProcess_74483322847285
// MI455X (gfx1250) — compile-verified
//
#include <hip/hip_runtime.h>
#include <cstdint>
#include <cstddef>

// Problem constants (from reference)
#define B_   16
#define A_   24564      // anchors (divisible by 4)
#define C_   81         // classes incl. background
#define CC_  80         // output classes
#define K_   200
#define TH_CONF 0.01f
#define TH_IOU  0.5f
#define EPS_    1e-8f

#define TILE  256                 // anchors per softmax tile
#define SPAN  2048                // anchors per softmax block
#define NPAD  24576               // A_ padded to 256*96
#define CHUNK 96                  // NPAD / 256

// ---------------- CDNA5 async global->LDS helpers (inline asm) ---------------
// GLOBAL_LOAD_ASYNC_TO_LDS_B128 (opcode 98, VGLOBAL GVS mode):
//   each lane copies 16 B from saddr + vaddr into LDS at vdst, tracked by ASYNCcnt.
__device__ __forceinline__ void async_ld_b128(uint32_t lds_byte_off,
                                              uint32_t g_byte_off,
                                              const void* sbase) {
  asm volatile("global_load_async_to_lds_b128 %0, %1, %2"
               :: "v"(lds_byte_off), "v"(g_byte_off), "s"(sbase)
               : "memory");
}
__device__ __forceinline__ void wait_async0() {
  asm volatile("s_wait_asynccnt 0" ::: "memory");
}
// Low 32 bits of a flat pointer into the LDS aperture == LDS byte offset.
__device__ __forceinline__ uint32_t lds_off(const void* p) {
  return (uint32_t)(uintptr_t)p;
}

// ---------------- CDNA5 Tensor Data Mover: 1D row DMA global->LDS ------------
typedef unsigned int u32x4 __attribute__((ext_vector_type(4)));
typedef int          i32x8 __attribute__((ext_vector_type(8)));
typedef int          i32x4 __attribute__((ext_vector_type(4)));

// Copy `ndwords` contiguous dwords from gsrc to the LDS byte address lds_byte with one TDM op.
// D# layout per CDNA5 ISA (1D tile: dims1..4 unused).
__device__ __forceinline__ void tdm_load_row_to_lds(uint32_t lds_byte,
                                                    const void* gsrc,
                                                    uint32_t ndwords) {
  uint64_t ga = (uint64_t)(uintptr_t)gsrc;
  u32x4 g0;
  g0.x = 1u;                                  // count=1 (valid), user mode
  g0.y = lds_byte;                            // lds_addr (bytes)
  g0.z = (uint32_t)ga;                        // global_addr[31:0]
  g0.w = (uint32_t)((ga >> 32) & 0x01FFFFFFu) // global_addr[56:32]
         | (2u << 30);                        // type = 2 ("image")
  i32x8 g1;
  g1[0] = (int)(2u << 16);                    // workgroup_mask=0, data_size=2 (4B)
  g1[1] = (int)((ndwords & 0xFFFFu) << 16);   // tensor_dim0[15:0] @ bits 63:48
  g1[2] = (int)((ndwords >> 16) | (1u << 16));// tensor_dim0[31:16] | tensor_dim1=1
  g1[3] = (int)((ndwords & 0xFFFFu) << 16);   // tile_dim0 @ bits 127:112
  g1[4] = 1;                                  // tile_dim1=1, tile_dim2=0
  g1[5] = (int)ndwords;                       // tensor_dim0_stride[31:0]
  g1[6] = 0;                                  // stride hi / tensor_dim1_stride
  g1[7] = 0;
  i32x4 g2 = {0, 0, 0, 0};                    // dims 2/3 unused
  i32x4 g3 = {0, 0, 0, 0};
#if __clang_major__ >= 23
  i32x8 g4 = {0, 0, 0, 0, 0, 0, 0, 0};
  __builtin_amdgcn_tensor_load_to_lds(g0, g1, g2, g3, g4, 0);
#else
  __builtin_amdgcn_tensor_load_to_lds(g0, g1, g2, g3, 0);
#endif
}

// ------------------------------ Kernel 1: decode -----------------------------
__global__ void decode_kernel(const float* __restrict__ loc,
                              const float* __restrict__ anchors,
                              float* __restrict__ boxes) {
  int idx = blockIdx.x * blockDim.x + threadIdx.x;
  if (idx >= B_ * A_) return;
  int a = idx % A_;
  float4 l  = ((const float4*)loc)[idx];
  float4 an = ((const float4*)anchors)[a];
  float cx = an.x + l.x * 0.1f * an.z;
  float cy = an.y + l.y * 0.1f * an.w;
  float w  = an.z * __expf(l.z * 0.2f);
  float h  = an.w * __expf(l.w * 0.2f);
  float4 o;
  o.x = cx - 0.5f * w; o.y = cy - 0.5f * h;
  o.z = cx + 0.5f * w; o.w = cy + 0.5f * h;
  ((float4*)boxes)[idx] = o;
}

// -------------------- Kernel 2: softmax + mask + transpose -------------------
// Tiles of TILE anchors x 81 classes are async-DMA'd into LDS, then each
// thread does a 3-pass softmax for its anchor and writes class-major scoresT.
__global__ void softmax_kernel(const float* __restrict__ conf,
                               float* __restrict__ scoresT) {
  extern __shared__ float smem[];       // TILE * C_ floats (82944 B)
  const int tid = threadIdx.x;
  const int b   = blockIdx.y;
  const int a_beg = blockIdx.x * SPAN;
  const int a_end = (a_beg + SPAN < A_) ? (a_beg + SPAN) : A_;

  for (int a0 = a_beg; a0 < a_end; a0 += TILE) {
    const int cnt  = ((A_ - a0) < TILE) ? (A_ - a0) : TILE;
    const int n128 = (cnt * C_) >> 2;   // cnt%4==0 -> cnt*81 dwords %4==0
    const float* gbase = conf + ((size_t)b * A_ + a0) * C_;
    const uint32_t lbase = lds_off(smem);
    for (int j = tid; j < n128; j += 256)
      async_ld_b128(lbase + (uint32_t)(j * 16), (uint32_t)(j * 16), gbase);
    wait_async0();
    __syncthreads();

    if (tid < cnt) {
      const float* row = smem + tid * C_;   // stride 81 (odd) -> bank friendly
      float m = row[0];
      #pragma unroll 4
      for (int c = 1; c < C_; ++c) m = fmaxf(m, row[c]);
      float sum = 0.f;
      #pragma unroll 4
      for (int c = 0; c < C_; ++c) sum += __expf(row[c] - m);
      float inv = 1.0f / sum;
      for (int c = 1; c < C_; ++c) {        // coalesced across tid per class
        float s = __expf(row[c] - m) * inv;
        float v = (s >= TH_CONF) ? s : -1.0f;
        scoresT[(size_t)(b * CC_ + (c - 1)) * A_ + a0 + tid] = v;
      }
    }
    __syncthreads();                        // before smem reuse
  }
}

// --------------------- Kernel 3: top-K per (batch,class) row -----------------
// Entire 24564-float row DMA'd into LDS by the Tensor Data Mover (one D# from
// wave 0). Per-thread chunk argmax cached in LDS; K iterations of wave32
// shuffle-argmax (+8-way cross-wave combine); only the winner rescans.
__global__ void topk_kernel(const float* __restrict__ scoresT,
                            float* __restrict__ topv,
                            int* __restrict__ topi) {
  extern __shared__ float row[];            // NPAD floats (98304 B)
  __shared__ float cmax[256];
  __shared__ int   cidx[256];
  __shared__ float wv[8];
  __shared__ int   ww[8];
  __shared__ int   bwin, bidx;

  const int tid = threadIdx.x;
  const int r   = blockIdx.x;               // r = b*80 + c
  const float* src = scoresT + (size_t)r * A_;

  if (tid < 32) {                           // one wave issues the TDM DMA
    tdm_load_row_to_lds(lds_off(row), src, A_);
    __builtin_amdgcn_s_wait_tensorcnt(0);
  }
  if (tid < NPAD - A_) row[A_ + tid] = -1e30f;   // pad tail
  __syncthreads();

  {
    const int base = tid * CHUNK;
    float m = row[base]; int mi = base;
    for (int i = 1; i < CHUNK; ++i) {
      float v = row[base + i];
      if (v > m) { m = v; mi = base + i; }
    }
    cmax[tid] = m; cidx[tid] = mi;
  }
  __syncthreads();

  for (int it = 0; it < K_; ++it) {
    // wave32 shuffle argmax (strict '>' keeps lowest index on ties)
    float v = cmax[tid];
    int   w = tid;
    #pragma unroll
    for (int off = 16; off > 0; off >>= 1) {
      float vo = __shfl_down(v, (unsigned)off, 32);
      int   wo = __shfl_down(w, (unsigned)off, 32);
      if (vo > v) { v = vo; w = wo; }
    }
    if ((tid & 31) == 0) { wv[tid >> 5] = v; ww[tid >> 5] = w; }
    __syncthreads();
    if (tid == 0) {
      float bv = wv[0]; int bw = ww[0];
      #pragma unroll
      for (int q = 1; q < 8; ++q)
        if (wv[q] > bv) { bv = wv[q]; bw = ww[q]; }
      bwin = bw; bidx = cidx[bw];
      topv[(size_t)r * K_ + it] = bv;
      topi[(size_t)r * K_ + it] = cidx[bw];
    }
    __syncthreads();
    if (tid == bwin) {                      // suppress winner, rescan chunk
      row[bidx] = -1e30f;
      const int base = tid * CHUNK;
      float m = row[base]; int mi = base;
      for (int i = 1; i < CHUNK; ++i) {
        float vv = row[base + i];
        if (vv > m) { m = vv; mi = base + i; }
      }
      cmax[tid] = m; cidx[tid] = mi;
    }
    __syncthreads();
  }
}

// -------------------------- Kernel 4: NMS + output ---------------------------
__global__ void nms_kernel(const float* __restrict__ boxes,
                           const float* __restrict__ topv,
                           const int* __restrict__ topi,
                           float* __restrict__ out) {
  __shared__ float X0[K_], Y0[K_], X1[K_], Y1[K_], AR[K_], SC[K_];
  __shared__ int KEEP[K_];
  const int tid = threadIdx.x;
  const int r   = blockIdx.x;
  const int b   = r / CC_;

  if (tid < K_) {
    int   idx = topi[(size_t)r * K_ + tid];
    float s   = topv[(size_t)r * K_ + tid];
    float4 bb = ((const float4*)boxes)[(size_t)b * A_ + idx];
    X0[tid] = bb.x; Y0[tid] = bb.y; X1[tid] = bb.z; Y1[tid] = bb.w;
    AR[tid] = fmaxf(bb.z - bb.x, 0.f) * fmaxf(bb.w - bb.y, 0.f);
    SC[tid] = s;
    KEEP[tid] = (s >= TH_CONF) ? 1 : 0;
  }
  __syncthreads();

  for (int i = 0; i < K_; ++i) {
    int ki = KEEP[i];
    if (ki && tid > i && tid < K_ && KEEP[tid]) {
      float lx = fmaxf(X0[i], X0[tid]);
      float ly = fmaxf(Y0[i], Y0[tid]);
      float rx = fminf(X1[i], X1[tid]);
      float ry = fminf(Y1[i], Y1[tid]);
      float inter = fmaxf(rx - lx, 0.f) * fmaxf(ry - ly, 0.f);
      float iou = inter / (AR[i] + AR[tid] - inter + EPS_);
      if (iou > TH_IOU) KEEP[tid] = 0;
    }
    __syncthreads();
  }

  if (tid < K_) {
    float m = KEEP[tid] ? 1.f : 0.f;
    size_t o = ((size_t)r * K_ + tid) * 5;
    out[o + 0] = X0[tid] * m;
    out[o + 1] = Y0[tid] * m;
    out[o + 2] = X1[tid] * m;
    out[o + 3] = Y1[tid] * m;
    out[o + 4] = SC[tid] * m;
  }
}

// ------------------------------- Host launcher -------------------------------
extern "C" void kernel_launch(void* const* d_in, const int* in_sizes, int n_in,
                              void* d_out, int out_size, void* d_ws, size_t ws_size,
                              hipStream_t stream) {
  const float* conf    = (const float*)d_in[0];
  const float* loc     = (const float*)d_in[1];
  const float* anchors = (const float*)d_in[2];
  float* out = (float*)d_out;

  // Workspace layout (floats): boxes | scoresT | topv | topi
  char* ws = (char*)d_ws;
  float* boxes   = (float*)ws;
  float* scoresT = (float*)(ws + (size_t)B_ * A_ * 4 * sizeof(float));
  float* topv    = scoresT + (size_t)B_ * CC_ * A_;
  int*   topi    = (int*)(topv + (size_t)B_ * CC_ * K_);

  // Allow >64KB dynamic LDS (gfx1250 WGP has 320KB). Idempotent, deterministic.
  (void)hipFuncSetAttribute((const void*)softmax_kernel,
                            hipFuncAttributeMaxDynamicSharedMemorySize,
                            TILE * C_ * (int)sizeof(float));
  (void)hipFuncSetAttribute((const void*)topk_kernel,
                            hipFuncAttributeMaxDynamicSharedMemorySize,
                            NPAD * (int)sizeof(float));

  decode_kernel<<<(B_ * A_ + 255) / 256, 256, 0, stream>>>(loc, anchors, boxes);

  dim3 g2((A_ + SPAN - 1) / SPAN, B_);
  softmax_kernel<<<g2, 256, TILE * C_ * sizeof(float), stream>>>(conf, scoresT);

  topk_kernel<<<B_ * CC_, 256, NPAD * sizeof(float), stream>>>(scoresT, topv, topi);

  nms_kernel<<<B_ * CC_, 256, 0, stream>>>(boxes, topv, topi, out);
}